// PhrasalLexemeAttention_77996606095876
// MI455X (gfx1250) — compile-verified
//
#include <hip/hip_runtime.h>
#include <hip/hip_bf16.h>

// ---------------------------------------------------------------------------
// PhrasalLexemeAttention for MI455X (gfx1250, wave32, WMMA).
// Shapes fixed by setup_inputs(): B=4, S=2048, D=512, D_MODEL=512.
// ---------------------------------------------------------------------------

typedef __attribute__((ext_vector_type(2))) float v2f;
typedef __attribute__((ext_vector_type(8))) float v8f;

constexpr int Bn  = 4;
constexpr int Sn  = 2048;
constexpr int Dn  = 512;   // input feature dim
constexpr int DMn = 512;   // d_model (projection out dim, also score divisor)
constexpr int BS  = Bn * Sn;          // 8192 rows
constexpr float NEG = -1e9f;

static __device__ __forceinline__ v8f wmma_f32(v2f a, v2f b, v8f c) {
  return __builtin_amdgcn_wmma_f32_16x16x4_f32(
      /*neg_a=*/false, a, /*neg_b=*/false, b,
      /*c_mod=*/(short)0, c, /*reuse_a=*/false, /*reuse_b=*/false);
}

// ---------------------------------------------------------------------------
// Kernel 1: Q = ctx @ Wq + bq ; K = ctx @ Wk + bk  via V_WMMA_F32_16X16X4_F32.
// Register-blocked: one wave computes a 32x32 region of BOTH Q and K
// (2 row-tiles x 2 col-tiles x 2 matrices = 8 accumulators), sharing the A
// (context) operand between the Q and K products and across column tiles.
// A layout (16x4 f32): lane L -> row M = L&15, half-wave g = L>>4 selects K
//   pair: VGPR0 = K = 2g, VGPR1 = K = 2g+1.
// B layout (4x16 f32): lane L -> col N = L&15, VGPRv holds K = 2g+v.
// C/D layout: VGPR r, lane L -> M = r + 8g, N = L&15.
// ---------------------------------------------------------------------------
__global__ __launch_bounds__(128) void qk_gemm(
    const float* __restrict__ ctx,
    const float* __restrict__ Wq, const float* __restrict__ bq,
    const float* __restrict__ Wk, const float* __restrict__ bk,
    float* __restrict__ Qo, float* __restrict__ Ko)
{
  const int lane = threadIdx.x & 31;
  const int wave = threadIdx.x >> 5;
  const int wid = blockIdx.x * 4 + wave;     // 4096 waves total
  constexpr int nColGrp = DMn / 32;          // 16 groups of 32 cols
  const int rowGrp = wid / nColGrp;          // 0..255 (32 rows each)
  const int colGrp = wid - rowGrp * nColGrp; // 0..15
  const int row0 = rowGrp * 32;
  const int col0 = colGrp * 32;

  const int m = lane & 15;    // A row / B col / D col
  const int g = lane >> 4;    // half-wave group

  const float* arow0 = ctx + (size_t)(row0 + m) * Dn + 2 * g;
  const float* arow1 = arow0 + (size_t)16 * Dn;
  const float* bqp0  = Wq + (size_t)(2 * g) * DMn + col0 + m;
  const float* bqp1  = bqp0 + 16;
  const float* bkp0  = Wk + (size_t)(2 * g) * DMn + col0 + m;
  const float* bkp1  = bkp0 + 16;

  v8f q00 = {}, q01 = {}, q10 = {}, q11 = {};
  v8f k00 = {}, k01 = {}, k10 = {}, k11 = {};

  for (int k0 = 0; k0 < Dn; k0 += 4) {
    const size_t o0 = (size_t)k0 * DMn;
    const size_t o1 = (size_t)(k0 + 1) * DMn;
    const v2f a0 = { arow0[k0], arow0[k0 + 1] };   // contiguous b64
    const v2f a1 = { arow1[k0], arow1[k0 + 1] };
    const v2f bq0 = { bqp0[o0], bqp0[o1] };
    const v2f bq1 = { bqp1[o0], bqp1[o1] };
    const v2f bk0 = { bkp0[o0], bkp0[o1] };
    const v2f bk1 = { bkp1[o0], bkp1[o1] };
    q00 = wmma_f32(a0, bq0, q00);
    q01 = wmma_f32(a0, bq1, q01);
    q10 = wmma_f32(a1, bq0, q10);
    q11 = wmma_f32(a1, bq1, q11);
    k00 = wmma_f32(a0, bk0, k00);
    k01 = wmma_f32(a0, bk1, k01);
    k10 = wmma_f32(a1, bk0, k10);
    k11 = wmma_f32(a1, bk1, k11);
  }

  const float bvq0 = bq[col0 + m];
  const float bvq1 = bq[col0 + 16 + m];
  const float bvk0 = bk[col0 + m];
  const float bvk1 = bk[col0 + 16 + m];

#pragma unroll
  for (int r = 0; r < 8; ++r) {
    const int mm = r + 8 * g;
    const size_t ro0 = (size_t)(row0 + mm) * DMn;
    const size_t ro1 = (size_t)(row0 + 16 + mm) * DMn;
    Qo[ro0 + col0 + m]      = q00[r] + bvq0;
    Qo[ro0 + col0 + 16 + m] = q01[r] + bvq1;
    Qo[ro1 + col0 + m]      = q10[r] + bvq0;
    Qo[ro1 + col0 + 16 + m] = q11[r] + bvq1;
    Ko[ro0 + col0 + m]      = k00[r] + bvk0;
    Ko[ro0 + col0 + 16 + m] = k01[r] + bvk1;
    Ko[ro1 + col0 + m]      = k10[r] + bvk0;
    Ko[ro1 + col0 + 16 + m] = k11[r] + bvk1;
  }
}

// ---------------------------------------------------------------------------
// Kernel 2: band dot products. One wave per (b,s):
//   ssuper[b,s] = (Q[b,s] . K[b,s+1]) / D_MODEL   (else -1e9)
//   ssub  [b,s] = (Q[b,s] . K[b,s-1]) / D_MODEL   (else -1e9)
// ---------------------------------------------------------------------------
__global__ __launch_bounds__(256) void band_dots(
    const float* __restrict__ Q, const float* __restrict__ K,
    float* __restrict__ ssub, float* __restrict__ ssuper)
{
  const int lane = threadIdx.x & 31;
  const int wid = blockIdx.x * 8 + (threadIdx.x >> 5);  // == b*S + s
  if (wid >= BS) return;
  const int s = wid & (Sn - 1);
  const bool hup = (s < Sn - 1);
  const bool hdn = (s > 0);

  const float* q   = Q + (size_t)wid * DMn;
  const float* kup = K + (size_t)wid * DMn + DMn;
  const float* kdn = K + (size_t)wid * DMn - DMn;

  float aup = 0.f, adn = 0.f;
  for (int j = lane; j < DMn; j += 32) {
    const float qv = q[j];
    if (hup) aup += qv * kup[j];
    if (hdn) adn += qv * kdn[j];
  }
#pragma unroll
  for (int off = 16; off; off >>= 1) {
    aup += __shfl_xor(aup, off, 32);
    adn += __shfl_xor(adn, off, 32);
  }
  if (lane == 0) {
    ssuper[wid] = hup ? aup * (1.0f / (float)DMn) : NEG;
    ssub[wid]   = hdn ? adn * (1.0f / (float)DMn) : NEG;
  }
}

// ---------------------------------------------------------------------------
// Kernel 3: per-batch softmax stats + superdiagonal log chain + prefix scan.
// One block per batch. Handles the general mask exactly like the reference:
// every row has S entries, non-band entries are -1e9 pre-softmax; z[s] is the
// common post-softmax value of the S-nb off-band entries.
// ---------------------------------------------------------------------------
__global__ __launch_bounds__(1024) void band_scan(
    const float* __restrict__ ssub, const float* __restrict__ ssuper,
    const int* __restrict__ amask,
    float* __restrict__ pplus, float* __restrict__ pminus,
    float* __restrict__ zrow, float* __restrict__ pasup,
    float* __restrict__ PS)
{
  __shared__ float buf0[Sn];
  __shared__ float buf1[Sn];
  const int b = blockIdx.x;
  const int tid = threadIdx.x;

  // Step 1: masked two-entry softmax per row.
  for (int s = tid; s < Sn; s += 1024) {
    const size_t base = (size_t)b * Sn + s;
    float vp = NEG, vm = NEG;
    int nb = 0;
    if (s < Sn - 1) { nb++; vp = (amask[base * Sn + s + 1] != 0) ? ssuper[base] : NEG; }
    if (s > 0)      { nb++; vm = (amask[base * Sn + s - 1] != 0) ? ssub[base]   : NEG; }
    const float mx = fmaxf(fmaxf(vp, vm), NEG);
    const float ep = (s < Sn - 1) ? __expf(vp - mx) : 0.f;
    const float em = (s > 0)      ? __expf(vm - mx) : 0.f;
    const float ez = __expf(NEG - mx);   // off-band entry value (0 unless row fully masked)
    const float denom = ep + em + (float)(Sn - nb) * ez;
    pplus[base]  = ep / denom;
    pminus[base] = em / denom;
    zrow[base]   = ez / denom;
  }
  __syncthreads();

  // Step 2: pa_super[s] = sqrt(p_plus[s]*p_minus[s+1] + 1e-9); L = log(pa+1e-9).
  for (int s = tid; s < Sn; s += 1024) {
    const size_t base = (size_t)b * Sn + s;
    float L = 0.f;
    if (s < Sn - 1) {
      const float pa = sqrtf(pplus[base] * pminus[base + 1] + 1e-9f);
      pasup[base] = pa;
      L = __logf(pa + 1e-9f);
    }
    buf0[s] = L;
  }
  __syncthreads();

  // Step 3: Hillis-Steele inclusive scan of L (double-buffered LDS).
  float* src = buf0;
  float* dst = buf1;
  for (int off = 1; off < Sn; off <<= 1) {
    for (int s = tid; s < Sn; s += 1024)
      dst[s] = src[s] + ((s >= off) ? src[s - off] : 0.f);
    __syncthreads();
    float* t = src; src = dst; dst = t;
  }
  // Exclusive prefix: PS[t] = sum_{u<t} L[u].
  for (int s = tid; s < Sn; s += 1024)
    PS[(size_t)b * Sn + s] = (s == 0) ? 0.f : src[s - 1];
}

// ---------------------------------------------------------------------------
// Kernel 4: materialize attn and pa (write-bound, float4 stores).
// One block per output row (b,s); 512 threads x 4 columns each.
//   attn[s,t] = t==s ? sqrt(z_s^2+1e-9) : exp(signed PS diff) + 1e-9
//   pa  [s,t] = band ? pa_super : sqrt(z_s*z_t + 1e-9)
// ---------------------------------------------------------------------------
__global__ __launch_bounds__(512) void fill_out(
    const float* __restrict__ PS, const float* __restrict__ zrow,
    const float* __restrict__ pasup,
    float* __restrict__ attn, float* __restrict__ pa)
{
  const int bs = blockIdx.x;           // b*S + s
  const int s  = bs & (Sn - 1);
  const int b  = bs >> 11;             // S = 2048
  const size_t rowBase  = (size_t)bs * Sn;
  const size_t batchOff = (size_t)b * Sn;

  const float PSs = PS[bs];            // uniform per block -> scalar loads
  const float zs  = zrow[bs];

  const int t0 = threadIdx.x * 4;
  const float4 pst4 = *(const float4*)(PS   + batchOff + t0);
  const float4 zt4  = *(const float4*)(zrow + batchOff + t0);
  const float pst[4] = { pst4.x, pst4.y, pst4.z, pst4.w };
  const float zt[4]  = { zt4.x,  zt4.y,  zt4.z,  zt4.w  };

  float av[4], pv[4];
#pragma unroll
  for (int j = 0; j < 4; ++j) {
    const int t = t0 + j;
    if (t == s) {
      const float d = sqrtf(zs * zs + 1e-9f);
      av[j] = d;
      pv[j] = d;
    } else {
      const float c = (t > s) ? (pst[j] - PSs) : (PSs - pst[j]);
      av[j] = __expf(c) + 1e-9f;
      const int dlt = t - s;
      if (dlt == 1)        pv[j] = pasup[bs];
      else if (dlt == -1)  pv[j] = pasup[bs - 1];
      else                 pv[j] = sqrtf(zs * zt[j] + 1e-9f);
    }
  }
  *(float4*)(attn + rowBase + t0) = make_float4(av[0], av[1], av[2], av[3]);
  *(float4*)(pa   + rowBase + t0) = make_float4(pv[0], pv[1], pv[2], pv[3]);
}

// ---------------------------------------------------------------------------
// Launch. Inputs: context[B,S,D] f32, attention_mask[B,S,S] i32,
// Wq[D,DM] f32, bq[DM] f32, Wk[D,DM] f32, bk[DM] f32.
// Output: attn[B,S,S] then pa[B,S,S], fp32, concatenated.
// ---------------------------------------------------------------------------
extern "C" void kernel_launch(void* const* d_in, const int* in_sizes, int n_in,
                              void* d_out, int out_size, void* d_ws, size_t ws_size,
                              hipStream_t stream)
{
  const float* ctx   = (const float*)d_in[0];
  const int*   amask = (const int*)  d_in[1];
  const float* Wq    = (const float*)d_in[2];
  const float* bq    = (const float*)d_in[3];
  const float* Wk    = (const float*)d_in[4];
  const float* bk    = (const float*)d_in[5];

  float* attn = (float*)d_out;                         // B*S*S
  float* pa   = attn + (size_t)Bn * Sn * Sn;           // B*S*S

  // Workspace carve-up (all float): Q, K, then 7 per-row arrays.
  float* Q      = (float*)d_ws;
  float* K      = Q      + (size_t)BS * DMn;
  float* ssub   = K      + (size_t)BS * DMn;
  float* ssuper = ssub   + BS;
  float* pplus  = ssuper + BS;
  float* pminus = pplus  + BS;
  float* zrow   = pminus + BS;
  float* pasup  = zrow   + BS;
  float* PS     = pasup  + BS;

  // 1) Projections via f32 WMMA: 4096 waves, each 32x32 of both Q and K.
  qk_gemm<<<1024, 128, 0, stream>>>(ctx, Wq, bq, Wk, bk, Q, K);
  // 2) Band dot products: one wave per (b,s).
  band_dots<<<1024, 256, 0, stream>>>(Q, K, ssub, ssuper);
  // 3) Softmax + log-chain prefix scan: one block per batch.
  band_scan<<<Bn, 1024, 0, stream>>>(ssub, ssuper, amask,
                                     pplus, pminus, zrow, pasup, PS);
  // 4) Materialize both outputs (write-bound).
  fill_out<<<BS, 512, 0, stream>>>(PS, zrow, pasup, attn, pa);
}